// GAT_29618094473881
// MI455X (gfx1250) — compile-verified
//
#include <hip/hip_runtime.h>
#include <math.h>

typedef __attribute__((ext_vector_type(2))) float v2f;
typedef __attribute__((ext_vector_type(8))) float v8f;

#define NEG_SLOPE 0.2f
#define SOFT_EPS  1e-16f

__device__ __forceinline__ float lrelu(float x) { return x > 0.f ? x : NEG_SLOPE * x; }
// order-preserving float<->int encoding so float max == int max
__device__ __forceinline__ int   enc_ord(float f) { int i = __float_as_int(f); return i >= 0 ? i : (i ^ 0x7FFFFFFF); }
__device__ __forceinline__ float dec_ord(int i)   { return __int_as_float(i >= 0 ? i : (i ^ 0x7FFFFFFF)); }

// ---------------------------------------------------------------------------
// fp32 WMMA GEMM: C[M,NC] = A[M,K] @ B[K,NC].  block = (32, NC/16), one wave
// per 16x16 C tile; B staged in LDS.  Uses V_WMMA_F32_16X16X4_F32.
// A frag (16x4 f32): v[0] = A[m, k0+2*half], v[1] = A[m, k0+2*half+1]
// B frag (4x16 f32): v[0] = B[k0+2*half, n], v[1] = B[k0+2*half+1, n]
// C frag (16x16)   : v[i] = C[i + 8*half, n]
// ---------------------------------------------------------------------------
__global__ void gat_gemm_wmma(const float* __restrict__ A, const float* __restrict__ Bg,
                              float* __restrict__ C, int M, int K, int NC) {
  extern __shared__ float Bs[];
  const int nthr = blockDim.x * blockDim.y;
  const int tid  = threadIdx.y * blockDim.x + threadIdx.x;
  for (int i = tid; i < K * NC; i += nthr) Bs[i] = Bg[i];
  __syncthreads();

  const int row0 = blockIdx.x * 16;
  const int col0 = threadIdx.y * 16;
  const int lane = threadIdx.x;
  const int mr   = lane & 15;
  const int half = lane >> 4;

  int arow = row0 + mr; if (arow > M - 1) arow = M - 1;
  const float* Ap = A + (size_t)arow * K;

  v8f acc = {};
  for (int k0 = 0; k0 < K; k0 += 4) {
    const int ka = k0 + 2 * half;
    v2f a, b;
    a.x = Ap[ka];
    a.y = Ap[ka + 1];
    b.x = Bs[ka * NC + col0 + mr];
    b.y = Bs[(ka + 1) * NC + col0 + mr];
    acc = __builtin_amdgcn_wmma_f32_16x16x4_f32(false, a, false, b, (short)0, acc,
                                                false, false);
  }
#pragma unroll
  for (int i = 0; i < 8; ++i) {
    const int r = row0 + i + 8 * half;
    if (r < M) C[(size_t)r * NC + col0 + mr] = acc[i];
  }
}

// ---------------------------------------------------------------------------
// Layer 1 (H=8, C=8): node prep -> edge max -> self denom -> edge sum ->
// edge aggregate -> finalize(+bias,+self,ELU)
// ---------------------------------------------------------------------------
__global__ void gat_prep1(const float* __restrict__ h1, const float* __restrict__ attS,
                          const float* __restrict__ attD, float* __restrict__ aS,
                          float* __restrict__ aD, int* __restrict__ menc,
                          float* __restrict__ denom, float* __restrict__ agg, int N) {
  const int t = blockIdx.x * blockDim.x + threadIdx.x;  // (n,h)
  if (t >= N * 8) return;
  const int h = t & 7;
  const float* hp = h1 + (size_t)t * 8;
  float s = 0.f, d = 0.f;
#pragma unroll
  for (int c = 0; c < 8; ++c) { s += hp[c] * attS[h * 8 + c]; d += hp[c] * attD[h * 8 + c]; }
  aS[t] = s; aD[t] = d;
  menc[t]  = enc_ord(lrelu(s + d));   // self-loop seeds the max
  denom[t] = 0.f;
  float* ap = agg + (size_t)t * 8;
#pragma unroll
  for (int c = 0; c < 8; ++c) ap[c] = 0.f;
}

__global__ void gat_emax1(const int* __restrict__ src, const int* __restrict__ dst,
                          const float* __restrict__ aS, const float* __restrict__ aD,
                          int* __restrict__ menc, int E) {
  const int e = blockIdx.x * blockDim.x + threadIdx.x;
  if (e >= E) return;
  const int s = src[e], d = dst[e];
#pragma unroll
  for (int h = 0; h < 8; ++h)
    atomicMax(&menc[d * 8 + h], enc_ord(lrelu(aS[s * 8 + h] + aD[d * 8 + h])));
}

__global__ void gat_dself1(const float* __restrict__ aS, const float* __restrict__ aD,
                           const int* __restrict__ menc, float* __restrict__ denom, int N) {
  const int t = blockIdx.x * blockDim.x + threadIdx.x;
  if (t >= N * 8) return;
  denom[t] = __expf(lrelu(aS[t] + aD[t]) - dec_ord(menc[t]));
}

__global__ void gat_esum1(const int* __restrict__ src, const int* __restrict__ dst,
                          const float* __restrict__ aS, const float* __restrict__ aD,
                          const int* __restrict__ menc, float* __restrict__ denom, int E) {
  const int e = blockIdx.x * blockDim.x + threadIdx.x;
  if (e >= E) return;
  const int s = src[e], d = dst[e];
#pragma unroll
  for (int h = 0; h < 8; ++h) {
    const float ev = lrelu(aS[s * 8 + h] + aD[d * 8 + h]);
    atomicAdd(&denom[d * 8 + h], __expf(ev - dec_ord(menc[d * 8 + h])));
  }
}

__global__ void gat_eagg1(const int* __restrict__ src, const int* __restrict__ dst,
                          const float* __restrict__ aS, const float* __restrict__ aD,
                          const int* __restrict__ menc, const float* __restrict__ denom,
                          const float* __restrict__ h1, float* __restrict__ agg, int E) {
  const int e = blockIdx.x * blockDim.x + threadIdx.x;
  if (e >= E) return;
  const int s = src[e], d = dst[e];
  __builtin_prefetch(h1 + (size_t)s * 64, 0, 1);
#pragma unroll
  for (int h = 0; h < 8; ++h) {
    const float ev = lrelu(aS[s * 8 + h] + aD[d * 8 + h]);
    const float alpha = __expf(ev - dec_ord(menc[d * 8 + h])) / (denom[d * 8 + h] + SOFT_EPS);
    const float4* hs = (const float4*)(h1 + (size_t)s * 64 + h * 8);
    const float4 p0 = hs[0], p1 = hs[1];
    float* o = agg + (size_t)d * 64 + h * 8;
    atomicAdd(o + 0, alpha * p0.x); atomicAdd(o + 1, alpha * p0.y);
    atomicAdd(o + 2, alpha * p0.z); atomicAdd(o + 3, alpha * p0.w);
    atomicAdd(o + 4, alpha * p1.x); atomicAdd(o + 5, alpha * p1.y);
    atomicAdd(o + 6, alpha * p1.z); atomicAdd(o + 7, alpha * p1.w);
  }
}

__global__ void gat_fin1(const float* __restrict__ agg, const float* __restrict__ h1,
                         const float* __restrict__ aS, const float* __restrict__ aD,
                         const int* __restrict__ menc, const float* __restrict__ denom,
                         const float* __restrict__ bias, float* __restrict__ hpost, int N) {
  const int t = blockIdx.x * blockDim.x + threadIdx.x;  // (n,h)
  if (t >= N * 8) return;
  const int h = t & 7;
  const float alpha =
      __expf(lrelu(aS[t] + aD[t]) - dec_ord(menc[t])) / (denom[t] + SOFT_EPS);
  const float* ap = agg + (size_t)t * 8;
  const float* hp = h1 + (size_t)t * 8;
  float* op = hpost + (size_t)t * 8;
#pragma unroll
  for (int c = 0; c < 8; ++c) {
    const float v = ap[c] + alpha * hp[c] + bias[h * 8 + c];
    op[c] = v > 0.f ? v : (__expf(v) - 1.f);   // ELU
  }
}

// ---------------------------------------------------------------------------
// Layer 2 (H=1, C=40, padded to 48 for WMMA n-tiles)
// ---------------------------------------------------------------------------
__global__ void gat_padw2(const float* __restrict__ W2, float* __restrict__ W2p) {
  const int t = blockIdx.x * blockDim.x + threadIdx.x;  // 64*48
  if (t >= 64 * 48) return;
  const int k = t / 48, c = t % 48;
  W2p[t] = (c < 40) ? W2[k * 40 + c] : 0.f;
}

__global__ void gat_prep2(const float* __restrict__ h2, const float* __restrict__ attS,
                          const float* __restrict__ attD, float* __restrict__ aS,
                          float* __restrict__ aD, int* __restrict__ menc,
                          float* __restrict__ denom, float* __restrict__ agg, int N) {
  const int n = blockIdx.x * blockDim.x + threadIdx.x;
  if (n >= N) return;
  const float* hp = h2 + (size_t)n * 48;
  float s = 0.f, d = 0.f;
#pragma unroll
  for (int c = 0; c < 40; ++c) { s += hp[c] * attS[c]; d += hp[c] * attD[c]; }
  aS[n] = s; aD[n] = d;
  menc[n]  = enc_ord(lrelu(s + d));
  denom[n] = 0.f;
  float* ap = agg + (size_t)n * 40;
#pragma unroll
  for (int c = 0; c < 40; ++c) ap[c] = 0.f;
}

__global__ void gat_emax2(const int* __restrict__ src, const int* __restrict__ dst,
                          const float* __restrict__ aS, const float* __restrict__ aD,
                          int* __restrict__ menc, int E) {
  const int e = blockIdx.x * blockDim.x + threadIdx.x;
  if (e >= E) return;
  atomicMax(&menc[dst[e]], enc_ord(lrelu(aS[src[e]] + aD[dst[e]])));
}

__global__ void gat_dself2(const float* __restrict__ aS, const float* __restrict__ aD,
                           const int* __restrict__ menc, float* __restrict__ denom, int N) {
  const int n = blockIdx.x * blockDim.x + threadIdx.x;
  if (n >= N) return;
  denom[n] = __expf(lrelu(aS[n] + aD[n]) - dec_ord(menc[n]));
}

__global__ void gat_esum2(const int* __restrict__ src, const int* __restrict__ dst,
                          const float* __restrict__ aS, const float* __restrict__ aD,
                          const int* __restrict__ menc, float* __restrict__ denom, int E) {
  const int e = blockIdx.x * blockDim.x + threadIdx.x;
  if (e >= E) return;
  const int s = src[e], d = dst[e];
  atomicAdd(&denom[d], __expf(lrelu(aS[s] + aD[d]) - dec_ord(menc[d])));
}

__global__ void gat_eagg2(const int* __restrict__ src, const int* __restrict__ dst,
                          const float* __restrict__ aS, const float* __restrict__ aD,
                          const int* __restrict__ menc, const float* __restrict__ denom,
                          const float* __restrict__ h2, float* __restrict__ agg, int E) {
  const int e = blockIdx.x * blockDim.x + threadIdx.x;
  if (e >= E) return;
  const int s = src[e], d = dst[e];
  __builtin_prefetch(h2 + (size_t)s * 48, 0, 1);
  const float alpha =
      __expf(lrelu(aS[s] + aD[d]) - dec_ord(menc[d])) / (denom[d] + SOFT_EPS);
  const float* hp = h2 + (size_t)s * 48;
  float* o = agg + (size_t)d * 40;
#pragma unroll
  for (int c = 0; c < 40; ++c) atomicAdd(o + c, alpha * hp[c]);
}

__global__ void gat_fin2(const float* __restrict__ agg, const float* __restrict__ h2,
                         const float* __restrict__ aS, const float* __restrict__ aD,
                         const int* __restrict__ menc, const float* __restrict__ denom,
                         const float* __restrict__ bias, float* __restrict__ out, int N) {
  const int n = blockIdx.x * blockDim.x + threadIdx.x;
  if (n >= N) return;
  const float alpha =
      __expf(lrelu(aS[n] + aD[n]) - dec_ord(menc[n])) / (denom[n] + SOFT_EPS);
  const float* ap = agg + (size_t)n * 40;
  const float* hp = h2 + (size_t)n * 48;
  float v[40];
  float mx = -INFINITY;
#pragma unroll
  for (int c = 0; c < 40; ++c) {
    v[c] = ap[c] + alpha * hp[c] + bias[c];   // mean over H=1 is identity
    mx = fmaxf(mx, v[c]);
  }
  float sum = 0.f;
#pragma unroll
  for (int c = 0; c < 40; ++c) sum += __expf(v[c] - mx);
  const float lse = mx + __logf(sum);
  float* op = out + (size_t)n * 40;
#pragma unroll
  for (int c = 0; c < 40; ++c) op[c] = v[c] - lse;
}

// ---------------------------------------------------------------------------
extern "C" void kernel_launch(void* const* d_in, const int* in_sizes, int n_in,
                              void* d_out, int out_size, void* d_ws, size_t ws_size,
                              hipStream_t stream) {
  const float* x   = (const float*)d_in[0];
  const int*   ei  = (const int*)d_in[1];
  const float* W1  = (const float*)d_in[2];
  const float* as1 = (const float*)d_in[3];
  const float* ad1 = (const float*)d_in[4];
  const float* b1  = (const float*)d_in[5];
  const float* W2  = (const float*)d_in[6];
  const float* as2 = (const float*)d_in[7];
  const float* ad2 = (const float*)d_in[8];
  const float* b2  = (const float*)d_in[9];
  float* out = (float*)d_out;

  const int Fin = in_sizes[2] / 64;        // W1: [F_in, 64]
  const int N   = in_sizes[0] / Fin;
  const int E   = in_sizes[1] / 2;         // edge_index: [2, E]
  const int* src = ei;
  const int* dst = ei + E;

  // workspace carve-out (floats)
  float* ws = (float*)d_ws;
  size_t o = 0;
  float* h1    = ws + o; o += (size_t)N * 64;
  float* hpost = ws + o; o += (size_t)N * 64;
  float* aS1   = ws + o; o += (size_t)N * 8;
  float* aD1   = ws + o; o += (size_t)N * 8;
  int*   m1    = (int*)(ws + o); o += (size_t)N * 8;
  float* dn1   = ws + o; o += (size_t)N * 8;
  float* agg1  = ws + o; o += (size_t)N * 64;
  float* h2    = ws + o; o += (size_t)N * 48;
  float* aS2   = ws + o; o += (size_t)N;
  float* aD2   = ws + o; o += (size_t)N;
  int*   m2    = (int*)(ws + o); o += (size_t)N;
  float* dn2   = ws + o; o += (size_t)N;
  float* agg2  = ws + o; o += (size_t)N * 40;
  float* W2p   = ws + o; o += (size_t)64 * 48;

  const int TB = 256;
  const int gNH = (N * 8 + TB - 1) / TB;
  const int gN  = (N + TB - 1) / TB;
  const int gE  = (E + TB - 1) / TB;
  const int gM  = (N + 15) / 16;

  // ---- layer 1 ----
  gat_gemm_wmma<<<gM, dim3(32, 4), (size_t)Fin * 64 * 4, stream>>>(x, W1, h1, N, Fin, 64);
  gat_prep1 <<<gNH, TB, 0, stream>>>(h1, as1, ad1, aS1, aD1, m1, dn1, agg1, N);
  gat_emax1 <<<gE,  TB, 0, stream>>>(src, dst, aS1, aD1, m1, E);
  gat_dself1<<<gNH, TB, 0, stream>>>(aS1, aD1, m1, dn1, N);
  gat_esum1 <<<gE,  TB, 0, stream>>>(src, dst, aS1, aD1, m1, dn1, E);
  gat_eagg1 <<<gE,  TB, 0, stream>>>(src, dst, aS1, aD1, m1, dn1, h1, agg1, E);
  gat_fin1  <<<gNH, TB, 0, stream>>>(agg1, h1, aS1, aD1, m1, dn1, b1, hpost, N);

  // ---- layer 2 ----
  gat_padw2 <<<(64 * 48 + TB - 1) / TB, TB, 0, stream>>>(W2, W2p);
  gat_gemm_wmma<<<gM, dim3(32, 3), (size_t)64 * 48 * 4, stream>>>(hpost, W2p, h2, N, 64, 48);
  gat_prep2 <<<gN, TB, 0, stream>>>(h2, as2, ad2, aS2, aD2, m2, dn2, agg2, N);
  gat_emax2 <<<gE, TB, 0, stream>>>(src, dst, aS2, aD2, m2, E);
  gat_dself2<<<gN, TB, 0, stream>>>(aS2, aD2, m2, dn2, N);
  gat_esum2 <<<gE, TB, 0, stream>>>(src, dst, aS2, aD2, m2, dn2, E);
  gat_eagg2 <<<gE, TB, 0, stream>>>(src, dst, aS2, aD2, m2, dn2, h2, agg2, E);
  gat_fin2  <<<gN, TB, 0, stream>>>(agg2, h2, aS2, aD2, m2, dn2, b2, out, N);

  (void)n_in; (void)out_size; (void)ws_size;
}